// KGEFactFilter_66460323938769
// MI455X (gfx1250) — compile-verified
//
#include <hip/hip_runtime.h>
#include <hip/hip_bf16.h>
#include <stdint.h>

// Problem constants (from the reference)
#define KF       1024      // facts per (b,s) row
#define DDIM     256       // embedding dim
#define NFACTS   1000000   // F
#define NEG_INF  (-1.0e9f)

#define THREADS  256            // 8 wave32 waves per workgroup
#define WAVES    (THREADS / 32)
#define FPW      (KF / WAVES)   // facts per wave = 128
#define KPT      (KF / THREADS) // keys per thread in select phases = 4
#define NSLOT    3              // async pipeline depth (facts in flight)

// 3 KB staging slot per fact: e_s @ +0, e_r @ +1024, e_o @ +2048 (bytes)
#define SLOT_FLOATS (3 * DDIM)

// Monotonic float -> uint key (larger key == larger float)
__device__ __forceinline__ unsigned order_key(float f) {
    unsigned u = __float_as_uint(f);
    return (u & 0x80000000u) ? ~u : (u | 0x80000000u);
}

struct Trip { int s, r, o; };

__device__ __forceinline__ Trip load_triple(const int* __restrict__ row_idx,
                                            const int* __restrict__ facts_idx,
                                            int j)
{
    int idx = min(max(row_idx[j], 0), NFACTS - 1);
    Trip tr;
    tr.s = facts_idx[idx * 3 + 0];
    tr.r = facts_idx[idx * 3 + 1];
    tr.o = facts_idx[idx * 3 + 2];
    return tr;
}

// Issue the 3-row gather for one fact as six per-lane async LDS copies.
// Each instruction moves 32 lanes x 16 B = 512 B; INST_OFFSET applies to both
// the LDS and the global address, so offset:512 covers each row's upper half.
__device__ __forceinline__ void issue_fact_async(
        const float* __restrict__ entity_emb,
        const float* __restrict__ rel_emb,
        Trip tr, unsigned lds_slot_byte, int lane)
{
    const unsigned loff = (unsigned)lane * 16u;
    uint64_t gs = (uint64_t)(uintptr_t)(entity_emb + (size_t)tr.s * DDIM) + loff;
    uint64_t gr = (uint64_t)(uintptr_t)(rel_emb    + (size_t)tr.r * DDIM) + loff;
    uint64_t go = (uint64_t)(uintptr_t)(entity_emb + (size_t)tr.o * DDIM) + loff;
    unsigned ls = lds_slot_byte + loff;           // e_s
    unsigned lr = lds_slot_byte + 1024u + loff;   // e_r
    unsigned lo = lds_slot_byte + 2048u + loff;   // e_o
    asm volatile(
        "global_load_async_to_lds_b128 %0, %3, off\n\t"
        "global_load_async_to_lds_b128 %0, %3, off offset:512\n\t"
        "global_load_async_to_lds_b128 %1, %4, off\n\t"
        "global_load_async_to_lds_b128 %1, %4, off offset:512\n\t"
        "global_load_async_to_lds_b128 %2, %5, off\n\t"
        "global_load_async_to_lds_b128 %2, %5, off offset:512"
        :
        : "v"(ls), "v"(lr), "v"(lo), "v"(gs), "v"(gr), "v"(go)
        : "memory");
}

__device__ __forceinline__ void wait_async_le(int facts_in_flight) {
    // 6 async ops per fact may remain outstanding per in-flight fact
    if (facts_in_flight == 2)      asm volatile("s_wait_asynccnt 0xC" ::: "memory");
    else if (facts_in_flight == 1) asm volatile("s_wait_asynccnt 0x6" ::: "memory");
    else                           asm volatile("s_wait_asynccnt 0x0" ::: "memory");
}

// One workgroup per row n. 8 waves; each wave scores 128 facts with a 3-deep
// async LDS gather pipeline, then the block radix-selects the top_k threshold
// over the 1024 scores and emits the keep mask.
__global__ __launch_bounds__(THREADS, 1)
void KGEFactFilter_66460323938769_kernel(
        const unsigned char* __restrict__ fact_success,  // [N*KF] bool bytes
        const int*           __restrict__ fact_item_idx, // [N*KF]
        const int*           __restrict__ facts_idx,     // [F*3] (s,r,o)
        const float*         __restrict__ entity_emb,    // [E*D]
        const float*         __restrict__ rel_emb,       // [R*D]
        const int*           __restrict__ top_k_p,       // [1]
        float*               __restrict__ out)           // [N*KF] 0/1
{
    __shared__ float    gbuf[WAVES][NSLOT][SLOT_FLOATS]; // 72 KB async staging
    __shared__ unsigned keys[KF];                        // 4 KB ordered keys
    __shared__ int      hist[256];                       // radix histogram
    __shared__ int      wave_tot[WAVES];
    __shared__ unsigned sh_prefix;
    __shared__ int      sh_krem;

    const int n    = blockIdx.x;
    const int t    = threadIdx.x;
    const int lane = t & 31;
    const int w    = t >> 5;

    // ---------------- Phase 1: async-pipelined gather + DistMult dot ------
    const int* row_idx = fact_item_idx + (size_t)n * KF;
    const int jbase = w * FPW;

    unsigned slot_byte[NSLOT];
    #pragma unroll
    for (int i = 0; i < NSLOT; ++i)
        slot_byte[i] = (unsigned)(uintptr_t)(&gbuf[w][i][0]);

    // Prologue: stage facts jbase, jbase+1 into slots 0,1; prefetch triple 2.
    {
        Trip t0 = load_triple(row_idx, facts_idx, jbase + 0);
        issue_fact_async(entity_emb, rel_emb, t0, slot_byte[0], lane);
        Trip t1 = load_triple(row_idx, facts_idx, jbase + 1);
        issue_fact_async(entity_emb, rel_emb, t1, slot_byte[1], lane);
    }
    Trip trip_cur = load_triple(row_idx, facts_idx, jbase + 2); // issued at jj=0
    Trip trip_nxt = trip_cur;

    int slot_c = 0;  // slot holding fact jj (consume)
    int slot_i = 2;  // slot receiving fact jj+2 (issue)

    for (int jj = 0; jj < FPW; ++jj) {
        const int j = jbase + jj;

        // Start the triple fetch for fact jj+3 (consumed next iteration).
        if (jj + 3 < FPW)
            trip_nxt = load_triple(row_idx, facts_idx, j + 3);

        // Issue fact jj+2 with the triple fetched one iteration ago.
        if (jj + 2 < FPW)
            issue_fact_async(entity_emb, rel_emb, trip_cur, slot_byte[slot_i], lane);

        const int in_flight = (jj + 2 < FPW) ? 2 : (jj + 1 < FPW) ? 1 : 0;
        wait_async_le(in_flight);

        // Consume the staged fact: lane covers dims [lane*8, lane*8+8).
        const float* slot = &gbuf[w][slot_c][0];
        const float4* ps = (const float4*)(slot +            lane * 8);
        const float4* pr = (const float4*)(slot + DDIM     + lane * 8);
        const float4* po = (const float4*)(slot + 2 * DDIM + lane * 8);
        const float4 a0 = ps[0], a1 = ps[1];
        const float4 b0 = pr[0], b1 = pr[1];
        const float4 c0 = po[0], c1 = po[1];

        float acc;
        acc = a0.x * b0.x * c0.x;
        acc = fmaf(a0.y * b0.y, c0.y, acc);
        acc = fmaf(a0.z * b0.z, c0.z, acc);
        acc = fmaf(a0.w * b0.w, c0.w, acc);
        acc = fmaf(a1.x * b1.x, c1.x, acc);
        acc = fmaf(a1.y * b1.y, c1.y, acc);
        acc = fmaf(a1.z * b1.z, c1.z, acc);
        acc = fmaf(a1.w * b1.w, c1.w, acc);

        #pragma unroll
        for (int off = 16; off > 0; off >>= 1)
            acc += __shfl_xor(acc, off, 32);

        if (lane == 0) {
            const unsigned char succ = fact_success[(size_t)n * KF + j];
            keys[j] = order_key(succ ? acc : NEG_INF);
        }

        trip_cur = trip_nxt;
        slot_c = (slot_c == NSLOT - 1) ? 0 : slot_c + 1;
        slot_i = (slot_i == NSLOT - 1) ? 0 : slot_i + 1;
    }
    __syncthreads();

    // ---------------- Phase 2: byte-radix select of the k-th largest ------
    const int k = top_k_p[0];
    unsigned mk[KPT];
    #pragma unroll
    for (int i = 0; i < KPT; ++i) mk[i] = keys[KPT * t + i];

    if (t == 0) { sh_prefix = 0u; sh_krem = k; }

    #pragma unroll
    for (int pass = 0; pass < 4; ++pass) {
        const int shift = 24 - pass * 8;

        hist[t] = 0;               // THREADS == 256 bins
        __syncthreads();

        const unsigned pfx   = sh_prefix;
        const unsigned pmask = (pass == 0) ? 0u : (0xFFFFFFFFu << (shift + 8));
        #pragma unroll
        for (int i = 0; i < KPT; ++i)
            if ((mk[i] & pmask) == (pfx & pmask))
                atomicAdd(&hist[(mk[i] >> shift) & 0xFF], 1);
        __syncthreads();

        if (t == 0) {
            const int krem = sh_krem;
            int cum = 0, b = 255;
            for (; b > 0; --b) {
                cum += hist[b];
                if (cum >= krem) break;
            }
            if (cum < krem) cum += hist[0];
            sh_krem   = krem - (cum - hist[b]);
            sh_prefix = pfx | ((unsigned)b << shift);
        }
        __syncthreads();
    }

    const unsigned T    = sh_prefix;
    const int      rrem = sh_krem;   // ties at T kept, earliest index first

    // ---------------- Phase 3: stable tie ranking + emit ------------------
    int local = 0;
    #pragma unroll
    for (int i = 0; i < KPT; ++i) local += (mk[i] == T);

    // wave32 inclusive scan of per-thread tie counts
    int incl = local;
    #pragma unroll
    for (int off = 1; off < 32; off <<= 1) {
        int v = __shfl_up(incl, off, 32);
        if (lane >= off) incl += v;
    }
    if (lane == 31) wave_tot[w] = incl;
    __syncthreads();
    if (t == 0) {
        int run = 0;
        for (int i = 0; i < WAVES; ++i) { int c = wave_tot[i]; wave_tot[i] = run; run += c; }
    }
    __syncthreads();

    int rank = wave_tot[w] + (incl - local);   // ties before my first key
    #pragma unroll
    for (int i = 0; i < KPT; ++i) {
        const int  j    = KPT * t + i;
        const bool eq   = (mk[i] == T);
        const bool keep = (mk[i] > T) || (eq && rank < rrem);
        if (eq) ++rank;
        const unsigned char succ = fact_success[(size_t)n * KF + j];
        out[(size_t)n * KF + j] = (succ && keep) ? 1.0f : 0.0f;
    }
}

extern "C" void kernel_launch(void* const* d_in, const int* in_sizes, int n_in,
                              void* d_out, int out_size, void* d_ws, size_t ws_size,
                              hipStream_t stream) {
    // setup_inputs() order:
    //   0 fact_success [B,S,KF] bool bytes
    //   1 fact_item_idx [N,KF] int32
    //   2 facts_idx [F,3] int32
    //   3 entity_emb [E,D] f32
    //   4 rel_emb [R,D] f32
    //   5 top_k scalar int
    const unsigned char* fact_success  = (const unsigned char*)d_in[0];
    const int*           fact_item_idx = (const int*)d_in[1];
    const int*           facts_idx     = (const int*)d_in[2];
    const float*         entity_emb    = (const float*)d_in[3];
    const float*         rel_emb       = (const float*)d_in[4];
    const int*           top_k        = (const int*)d_in[5];
    float*               out          = (float*)d_out;

    const int N = in_sizes[1] / KF;   // B*S = 256 rows

    hipLaunchKernelGGL(KGEFactFilter_66460323938769_kernel,
                       dim3(N), dim3(THREADS), 0, stream,
                       fact_success, fact_item_idx, facts_idx,
                       entity_emb, rel_emb, top_k, out);
}